// RNN_42417097015547
// MI455X (gfx1250) — compile-verified
//
#include <hip/hip_runtime.h>
#include <math.h>

typedef __attribute__((ext_vector_type(2))) float v2f;
typedef __attribute__((ext_vector_type(8))) float v8f;

#define NTHREADS 256
#define NWAVES 8

// LDS row strides (in floats); odd-ish to avoid bank conflicts
#define AS 257   // activation ping-pong buffers, max K = 256
#define ZS 1027  // gate pre-activation buffer, max 4H = 1024

struct SharedMem {
    float bufA[16 * AS];
    float bufB[16 * AS];
    float z[16 * ZS];
};

// One wave computes 16x16 output tiles of  Z[16,Nout] = A[16,K] @ W[Nout,K]^T
// using V_WMMA_F32_16X16X4_F32, striding K by 4.
// Wave index is read via readfirstlane so the tile loop is a scalar (SALU)
// loop: EXEC is provably all-ones at every WMMA (ISA requirement).
__device__ __forceinline__ void gemm_wmma(const float* __restrict__ A_lds,  // [16][AS] in LDS
                                          const float* __restrict__ W,     // [Nout][K] row-major, global
                                          float* __restrict__ z_lds,       // [16][ZS] in LDS
                                          int K, int Nout)
{
    const int lane  = threadIdx.x & 31;
    const int wave  = __builtin_amdgcn_readfirstlane(threadIdx.x >> 5);  // scalar
    const int lcol  = lane & 15;   // M for A, N for B, N for C/D
    const int lhalf = lane >> 4;   // K-pair select for A/B, row-half for C/D

    for (int nt = wave; nt < (Nout >> 4); nt += NWAVES) {
        const int n0 = nt << 4;    // scalar
        const int n  = n0 + lcol;
        v8f acc = {};
        for (int k = 0; k < K; k += 4) {
            const int kk = k + lhalf * 2;
            v2f a, b;
            // A 16x4: VGPR0 = K = k + 2*lhalf, VGPR1 = K+1 ; lane&15 = M
            a.x = (kk     < K) ? A_lds[lcol * AS + kk    ] : 0.0f;
            a.y = (kk + 1 < K) ? A_lds[lcol * AS + kk + 1] : 0.0f;
            // B 4x16 = W^T: element (K=kk, N=n) = W[n*K + kk] ; lane&15 = N
            b.x = (kk     < K) ? W[n * K + kk    ] : 0.0f;
            b.y = (kk + 1 < K) ? W[n * K + kk + 1] : 0.0f;
            acc = __builtin_amdgcn_wmma_f32_16x16x4_f32(
                false, a, false, b, (short)0, acc, false, false);
        }
        // C/D: VGPR r -> row (lhalf*8 + r), col = n
        #pragma unroll
        for (int r = 0; r < 8; ++r)
            z_lds[(lhalf * 8 + r) * ZS + n] = acc[r];
    }
}

__device__ __forceinline__ float sigmoidf_(float x) { return 1.0f / (1.0f + expf(-x)); }

// LSTM cell with h0=c0=0:  c = sig(i)*tanh(g), h = sig(o)*tanh(c), out = tanh(h)
__device__ __forceinline__ void lstm_gates(const float* __restrict__ z,
                                           const float* __restrict__ bias,
                                           float* __restrict__ out, int H)
{
    for (int idx = threadIdx.x; idx < 16 * H; idx += NTHREADS) {
        int r = idx / H, hi = idx - r * H;
        const float* zr = z + r * ZS;
        float ig = zr[hi]         + bias[hi];
        float gg = zr[2 * H + hi] + bias[2 * H + hi];
        float og = zr[3 * H + hi] + bias[3 * H + hi];
        float c  = sigmoidf_(ig) * tanhf(gg);
        float h  = sigmoidf_(og) * tanhf(c);
        out[r * AS + hi] = tanhf(h);
    }
}

__device__ __forceinline__ void fc_relu(const float* __restrict__ z,
                                        const float* __restrict__ bias,
                                        float* __restrict__ out, int N)
{
    for (int idx = threadIdx.x; idx < 16 * N; idx += NTHREADS) {
        int r = idx / N, n = idx - r * N;
        float v = z[r * ZS + n] + bias[n];
        out[r * AS + n] = v > 0.0f ? v : 0.0f;
    }
}

__global__ __launch_bounds__(NTHREADS) void rnn_fused(
    const float* __restrict__ x,
    const float* __restrict__ w1i, const float* __restrict__ b1,
    const float* __restrict__ w2i, const float* __restrict__ b2,
    const float* __restrict__ w3i, const float* __restrict__ b3,
    const float* __restrict__ w4i, const float* __restrict__ b4,
    const float* __restrict__ f1w, const float* __restrict__ f1b,
    const float* __restrict__ f2w, const float* __restrict__ f2b,
    const float* __restrict__ f3w, const float* __restrict__ f3b,
    const float* __restrict__ f4w, const float* __restrict__ f4b,
    float* __restrict__ out)
{
    __shared__ SharedMem s;
    const int M0 = blockIdx.x * 16;   // 16 batch rows per workgroup

    // Stage x[t=0] tile [16 x 87] into LDS (only timestep 0 matters).
    for (int idx = threadIdx.x; idx < 16 * 87; idx += NTHREADS) {
        int r = idx / 87, k = idx - r * 87;
        s.bufA[r * AS + k] = x[(M0 + r) * 87 + k];
    }
    __syncthreads();

    // LSTM 1: 87 -> H=256
    gemm_wmma(s.bufA, w1i, s.z, 87, 1024);
    __syncthreads();
    lstm_gates(s.z, b1, s.bufB, 256);
    __syncthreads();

    // LSTM 2: 256 -> H=128
    gemm_wmma(s.bufB, w2i, s.z, 256, 512);
    __syncthreads();
    lstm_gates(s.z, b2, s.bufA, 128);
    __syncthreads();

    // LSTM 3: 128 -> H=64
    gemm_wmma(s.bufA, w3i, s.z, 128, 256);
    __syncthreads();
    lstm_gates(s.z, b3, s.bufB, 64);
    __syncthreads();

    // LSTM 4: 64 -> H=32
    gemm_wmma(s.bufB, w4i, s.z, 64, 128);
    __syncthreads();
    lstm_gates(s.z, b4, s.bufA, 32);
    __syncthreads();

    // FC1: 32 -> 128, relu
    gemm_wmma(s.bufA, f1w, s.z, 32, 128);
    __syncthreads();
    fc_relu(s.z, f1b, s.bufB, 128);
    __syncthreads();

    // FC2: 128 -> 64, relu
    gemm_wmma(s.bufB, f2w, s.z, 128, 64);
    __syncthreads();
    fc_relu(s.z, f2b, s.bufA, 64);
    __syncthreads();

    // FC3: 64 -> 32, relu
    gemm_wmma(s.bufA, f3w, s.z, 64, 32);
    __syncthreads();
    fc_relu(s.z, f3b, s.bufB, 32);
    __syncthreads();

    // FC4: 32 -> 3 (tiny; plain VALU dot products)
    if (threadIdx.x < 48) {
        int r = threadIdx.x / 3, j = threadIdx.x - r * 3;
        float acc = f4b[j];
        #pragma unroll
        for (int k = 0; k < 32; ++k)
            acc += s.bufB[r * AS + k] * f4w[j * 32 + k];
        out[(M0 + r) * 3 + j] = acc;
    }
}

extern "C" void kernel_launch(void* const* d_in, const int* in_sizes, int n_in,
                              void* d_out, int out_size, void* d_ws, size_t ws_size,
                              hipStream_t stream)
{
    (void)in_sizes; (void)n_in; (void)out_size; (void)d_ws; (void)ws_size;
    const float* x   = (const float*)d_in[0];
    const float* w1i = (const float*)d_in[1];   // w1h (d_in[2]) unused: h0 = 0
    const float* b1  = (const float*)d_in[3];
    const float* w2i = (const float*)d_in[4];
    const float* b2  = (const float*)d_in[6];
    const float* w3i = (const float*)d_in[7];
    const float* b3  = (const float*)d_in[9];
    const float* w4i = (const float*)d_in[10];
    const float* b4  = (const float*)d_in[12];
    const float* f1w = (const float*)d_in[13];
    const float* f1b = (const float*)d_in[14];
    const float* f2w = (const float*)d_in[15];
    const float* f2b = (const float*)d_in[16];
    const float* f3w = (const float*)d_in[17];
    const float* f3b = (const float*)d_in[18];
    const float* f4w = (const float*)d_in[19];
    const float* f4b = (const float*)d_in[20];

    rnn_fused<<<dim3(16), dim3(NTHREADS), 0, stream>>>(
        x, w1i, b1, w2i, b2, w3i, b3, w4i, b4,
        f1w, f1b, f2w, f2b, f3w, f3b, f4w, f4b,
        (float*)d_out);
}